// GCNConvNet_44152263803032
// MI455X (gfx1250) — compile-verified
//
#include <hip/hip_runtime.h>
#include <hip/hip_bf16.h>

typedef __attribute__((ext_vector_type(2))) float v2f;
typedef __attribute__((ext_vector_type(8))) float v8f;

// ---------------------------------------------------------------------------
// Degree: deg[col[e]] += 1
__global__ void deg_kernel(const int* __restrict__ col, float* __restrict__ deg, int E) {
    int e = blockIdx.x * blockDim.x + threadIdx.x;
    if (e < E) atomicAdd(&deg[col[e]], 1.0f);
}

// dinv in place: deg[i] = deg[i] > 0 ? rsqrt(deg[i]) : 0
__global__ void dinv_kernel(float* __restrict__ deg, int N) {
    int i = blockIdx.x * blockDim.x + threadIdx.x;
    if (i < N) {
        float d = deg[i];
        deg[i] = (d > 0.0f) ? rsqrtf(d) : 0.0f;
    }
}

// norm[e] = dinv[row[e]] * dinv[col[e]]
__global__ void norm_kernel(const int* __restrict__ row, const int* __restrict__ col,
                            const float* __restrict__ dinv, float* __restrict__ norm, int E) {
    int e = blockIdx.x * blockDim.x + threadIdx.x;
    if (e < E) norm[e] = dinv[row[e]] * dinv[col[e]];
}

// ---------------------------------------------------------------------------
// Edge aggregation: agg[col[e]][:] += h[row[e]][:] * norm[e]
// One wave per edge (32 lanes x float2 = 64 channels). The edge id is wave-
// uniform: force it into an SGPR with readfirstlane so row/col/norm become
// scalar (SMEM) loads instead of 3 broadcast VMEM loads per wave. The h-row
// gather is one coalesced 256B global_load_b64 per wave; the 64 atomic adds
// land in the 192MB L2 (h/agg are only 25.6MB each).
__global__ void agg_kernel(const float* __restrict__ h, const float* __restrict__ norm,
                           const int* __restrict__ row, const int* __restrict__ col,
                           float* __restrict__ agg, int E) {
    long t = (long)blockIdx.x * blockDim.x + threadIdx.x;
    int e = __builtin_amdgcn_readfirstlane((int)(t >> 5));   // wave-uniform edge id
    if (e >= E) return;
    int c = (int)(t & 31) * 2;
    int r  = row[e];     // scalar load
    int cl = col[e];     // scalar load
    float w = norm[e];   // scalar load
    float2 hv = *(const float2*)(h + (long)r * 64 + c);
    float* ap = agg + (long)cl * 64 + c;
    atomicAdd(ap,     hv.x * w);
    atomicAdd(ap + 1, hv.y * w);
}

// ---------------------------------------------------------------------------
// Fused WMMA GEMM (f32, exact):  C = act( (s1*A1 + s2*A2) * B + bias )
//   A1,A2 : [M,K] row-major (A2 enabled by template flag)
//   B     : transB ? [Ncols,K] : [K,Ncols]   (Ncols <= 64, K % 4 == 0)
//   C1,C2 : [M,Ncols] row-major (C2 optional mirror write)
// Block: 128 threads = 4 waves; wave w -> rows [blk*64 + w*16, +16), all Ncols.
// Template NT = number of 16-wide n-tiles -> branch-free unrolled WMMA chain.
// EXEC all-ones throughout (no early returns); row OOB handled by zero scales.
template<int NT, bool HAS_A2>
__launch_bounds__(128)
__global__ void gemm_wmma_f32(const float* __restrict__ A1, const float* __restrict__ A2,
                              float s1, float s2,
                              const float* __restrict__ B, int transB,
                              const float* __restrict__ bias, int relu,
                              float* __restrict__ C1, float* __restrict__ C2,
                              int M, int K, int Ncols) {
    extern __shared__ float Bs[];  // [K][64], columns >= Ncols zero-padded

    const int tid  = threadIdx.x;
    const int lane = tid & 31;
    const int wave = tid >> 5;
    const int half = lane >> 4;   // 0: lanes 0-15, 1: lanes 16-31
    const int lm   = lane & 15;

    // Cooperative load of B into LDS as [K][64] with zero padding.
    for (int idx = tid; idx < K * 64; idx += 128) {
        int k = idx >> 6;
        int n = idx & 63;
        float v = 0.0f;
        if (n < Ncols) v = transB ? B[(long)n * K + k] : B[(long)k * Ncols + n];
        Bs[idx] = v;
    }
    __syncthreads();

    // Row this lane supplies for the A fragment
    // (A 16x4 layout: lanes 0-15 hold K={0,1}, lanes 16-31 hold K={2,3}).
    long m = (long)blockIdx.x * 64 + wave * 16 + lm;
    bool mv = (m < (long)M);
    // OOB rows: read (valid, finite) row 0 and scale by 0 -> no per-iter predication.
    const float* a1p = A1 + (mv ? m : 0) * (long)K;
    const float* a2p = HAS_A2 ? (A2 + (mv ? m : 0) * (long)K) : nullptr;
    const float es1 = mv ? s1 : 0.0f;
    const float es2 = mv ? s2 : 0.0f;

    v8f acc[NT];
    #pragma unroll
    for (int t = 0; t < NT; ++t) acc[t] = (v8f){};

    #pragma unroll 4
    for (int k0 = 0; k0 < K; k0 += 4) {
        const int ka = k0 + 2 * half;      // even -> 8B-aligned float2 loads
        float2 a1 = *(const float2*)(a1p + ka);
        v2f a;
        if (HAS_A2) {
            float2 a2 = *(const float2*)(a2p + ka);
            a.x = es1 * a1.x + es2 * a2.x;
            a.y = es1 * a1.y + es2 * a2.y;
        } else {
            a.x = es1 * a1.x;
            a.y = es1 * a1.y;
        }
        // B fragment for n-tile t: VGPR r holds K = ka + r, N = t*16 + lm.
        #pragma unroll
        for (int t = 0; t < NT; ++t) {
            v2f b;
            b.x = Bs[ka * 64 + t * 16 + lm];
            b.y = Bs[(ka + 1) * 64 + t * 16 + lm];
            acc[t] = __builtin_amdgcn_wmma_f32_16x16x4_f32(
                false, a, false, b, (short)0, acc[t], false, false);
        }
    }

    // Store: C/D layout -> lane half selects M+8, VGPR index r is row offset.
    #pragma unroll
    for (int t = 0; t < NT; ++t) {
        int n = t * 16 + lm;
        float bv = (bias != nullptr && n < Ncols) ? bias[n] : 0.0f;
        #pragma unroll
        for (int r = 0; r < 8; ++r) {
            long mm = (long)blockIdx.x * 64 + wave * 16 + r + 8 * half;
            if (mm < (long)M && n < Ncols) {
                float v = acc[t][r] + bv;
                if (relu) v = fmaxf(v, 0.0f);
                C1[mm * Ncols + n] = v;
                if (C2) C2[mm * Ncols + n] = v;
            }
        }
    }
}

// ---------------------------------------------------------------------------
extern "C" void kernel_launch(void* const* d_in, const int* in_sizes, int n_in,
                              void* d_out, int out_size, void* d_ws, size_t ws_size,
                              hipStream_t stream) {
    const float* x      = (const float*)d_in[0];
    const int*   ei     = (const int*)d_in[1];   // [2, E]
    const float* fc1_w  = (const float*)d_in[3]; // [NH, INC]
    const float* fc1_b  = (const float*)d_in[4]; // [NH]
    const float* conv_w = (const float*)d_in[5]; // [NL, NH, NH]
    const float* fc2_w  = (const float*)d_in[6]; // [OUTC, NH]
    const float* fc2_b  = (const float*)d_in[7]; // [OUTC]
    float* out = (float*)d_out;

    const int NH   = in_sizes[4];               // 64
    const int INC  = in_sizes[3] / NH;          // 128
    const int OUTC = in_sizes[7];               // 40
    const int NL   = in_sizes[5] / (NH * NH);   // 4
    const int N    = in_sizes[0] / INC;         // 100000
    const int E    = in_sizes[1] / 2;           // 1600000
    const float ALPHA = 0.1f;

    const int* rowp = ei;
    const int* colp = ei + E;

    // Workspace layout
    char* ws = (char*)d_ws;
    size_t o = 0;
    float* deg  = (float*)(ws + o); o += (size_t)N * 4;        o = (o + 255) & ~(size_t)255;
    float* norm = (float*)(ws + o); o += (size_t)E * 4;        o = (o + 255) & ~(size_t)255;
    float* h    = (float*)(ws + o); o += (size_t)N * NH * 4;   o = (o + 255) & ~(size_t)255;
    float* h0   = (float*)(ws + o); o += (size_t)N * NH * 4;   o = (o + 255) & ~(size_t)255;
    float* agg  = (float*)(ws + o);

    // 1) degree -> dinv -> per-edge norm
    hipMemsetAsync(deg, 0, (size_t)N * 4, stream);
    deg_kernel<<<(E + 255) / 256, 256, 0, stream>>>(colp, deg, E);
    dinv_kernel<<<(N + 255) / 256, 256, 0, stream>>>(deg, N);
    norm_kernel<<<(E + 255) / 256, 256, 0, stream>>>(rowp, colp, deg, norm, E);

    const int mblocks = (N + 63) / 64;

    // 2) h = h0 = relu(x @ fc1_w^T + fc1_b)          (NT=4 tiles, no A2)
    gemm_wmma_f32<4, false><<<mblocks, 128, (size_t)INC * 64 * 4, stream>>>(
        x, nullptr, 1.0f, 0.0f, fc1_w, /*transB=*/1, fc1_b, /*relu=*/1,
        h, h0, N, INC, NH);

    // 3) GCN2 layers
    long aggThreads = (long)E * 32;
    int aggBlocks = (int)((aggThreads + 255) / 256);
    for (int i = 0; i < NL; ++i) {
        hipMemsetAsync(agg, 0, (size_t)N * NH * 4, stream);
        agg_kernel<<<aggBlocks, 256, 0, stream>>>(h, norm, rowp, colp, agg, E);
        // h = relu( (0.9*agg + 0.1*h0) @ conv_w[i] )  (NT=4 tiles, with A2 mix)
        gemm_wmma_f32<4, true><<<mblocks, 128, (size_t)NH * 64 * 4, stream>>>(
            agg, h0, 1.0f - ALPHA, ALPHA, conv_w + (size_t)i * NH * NH, /*transB=*/0,
            nullptr, /*relu=*/1, h, nullptr, N, NH, NH);
    }

    // 4) out = h @ fc2_w^T + fc2_b                   (NT=3 tiles covers 40 cols)
    gemm_wmma_f32<3, false><<<mblocks, 128, (size_t)NH * 64 * 4, stream>>>(
        h, nullptr, 1.0f, 0.0f, fc2_w, /*transB=*/1, fc2_b, /*relu=*/0,
        out, nullptr, N, NH, OUTC);
}